// PhysicalLayer_14628658610555
// MI455X (gfx1250) — compile-verified
//
#include <hip/hip_runtime.h>
#include <math.h>

// ---------------------------------------------------------------------------
// MI455X (gfx1250) implementation.
//
// Pipeline (complex matrices stored as separate re/im f32 planes in d_ws):
//   field = B1 * mask_real * exp(i*mask_param)                (500x500)
//   FE = Dpad^T @ (field @ Dpad)          pad-aware fft2(E1)  (1000x1000)
//   FQ = D1000 @ (Q1 @ D1000)             fft2(Q1)            (1000x1000)
//   P  = FE .* FQ
//   out = 1e-6 * Einv^T @ (P @ Einv)      cropped ifft2+shift (500x500)
//   G  = D500 @ out @ D500                fft2(out)           (500x500)
//   per emitter e: W[k] = sum_l G[k,l]*exp(i*s*gshift[k,l]*xf_e)*e^{2pi i c l/500}
//                  intensity_e = (1/500^3) * sum_k |W[k]|^2       (Parseval)
//   scatter 31x31 psf patches * intensity into canvas, then noise pipeline.
//
// All GEMMs use V_WMMA_F32_16X16X4_F32 (f32 matrix path: DFT needs f32).
// Whole working set (~78 MB) is L2-resident (192 MB), so GEMMs are
// compute-bound even with direct global operand fetch.
// ---------------------------------------------------------------------------

#define TWO_PI_F 6.28318530717958647692f

typedef float v2f __attribute__((ext_vector_type(2)));
typedef float v8f __attribute__((ext_vector_type(8)));

__device__ __forceinline__ v8f wmma4_f32(v2f a, v2f b, v8f c) {
  // 8-arg form: (neg_a, A, neg_b, B, c_mod, C, reuse_a, reuse_b)
  return __builtin_amdgcn_wmma_f32_16x16x4_f32(false, a, false, b,
                                               (short)0, c, false, false);
}

// ---------------------------------------------------------------------------
// Complex GEMM: C = alpha * A(MxK) * B(KxN), row-major, K % 4 == 0.
// One 16x32 C tile per wave32 (two 16x16 WMMA tiles sharing the A fragment),
// 4 waves -> 32x64 macro tile per 128-thread block.
//
// Branch-free inner loop: out-of-range lanes CLAMP their load address to a
// valid row/col instead of masking.  WMMA dataflow localizes A-row m to C-row
// m and B-col n to C-col n, so clamped (finite) garbage only reaches C
// entries that the bounds-guarded store never writes.  No EXEC manipulation
// inside the K loop -> WMMA-issue-bound.
//
// Fragment layouts per CDNA5 ISA 7.12.2:
//   A 16x4 f32 : lane m=L%16, half=L/16 -> {A[m][k0+2h], A[m][k0+2h+1]} (b64)
//   B 4x16 f32 : lane n=L%16, half     -> {B[k0+2h][n], B[k0+2h+1][n]}
//   C 16x16    : acc[r] = C[tm + r + 8*half][tn + n]
// ---------------------------------------------------------------------------
__global__ __launch_bounds__(128)
void cgemm_wmma_f32(const float* __restrict__ Ar, const float* __restrict__ Ai,
                    const float* __restrict__ Br, const float* __restrict__ Bi,
                    float* __restrict__ Cr, float* __restrict__ Ci,
                    int M, int N, int K, float alpha)
{
  const int lane = threadIdx.x & 31;
  const int wave = threadIdx.x >> 5;
  const int half = lane >> 4;
  const int l16  = lane & 15;

  const int tm = blockIdx.y * 32 + (wave >> 1) * 16;
  const int tn = blockIdx.x * 64 + (wave & 1) * 32;

  // Clamped fetch coordinates (branch-free; see note above).
  const int am  = min(tm + l16, M - 1);
  const int bn0 = min(tn + l16, N - 1);
  const int bn1 = min(tn + 16 + l16, N - 1);

  // A fragment: contiguous pair, 8-byte aligned (K even) -> b64 loads.
  const v2f* ar2 = (const v2f*)(Ar + (size_t)am * (size_t)K) + half;
  const v2f* ai2 = (const v2f*)(Ai + (size_t)am * (size_t)K) + half;

  // B running pointers: row ka = k0 + 2*half; .y partner at +N.
  const size_t rowsz = (size_t)N;
  const float* pbr0 = Br + (size_t)(2 * half) * rowsz + bn0;
  const float* pbi0 = Bi + (size_t)(2 * half) * rowsz + bn0;
  const float* pbr1 = Br + (size_t)(2 * half) * rowsz + bn1;
  const float* pbi1 = Bi + (size_t)(2 * half) * rowsz + bn1;
  const size_t bstep = 4 * rowsz;

  v8f accR0 = {}, accI0 = {}, accR1 = {}, accI1 = {};

  for (int k0 = 0; k0 < K; k0 += 4) {
    const v2f a_r = *ar2;  ar2 += 2;
    const v2f a_i = *ai2;  ai2 += 2;

    v2f br0, bi0, br1, bi1, nbi0, nbi1;
    br0.x = pbr0[0];  br0.y = pbr0[rowsz];  pbr0 += bstep;
    bi0.x = pbi0[0];  bi0.y = pbi0[rowsz];  pbi0 += bstep;
    br1.x = pbr1[0];  br1.y = pbr1[rowsz];  pbr1 += bstep;
    bi1.x = pbi1[0];  bi1.y = pbi1[rowsz];  pbi1 += bstep;

    nbi0.x = -bi0.x;  nbi0.y = -bi0.y;
    nbi1.x = -bi1.x;  nbi1.y = -bi1.y;

    accR0 = wmma4_f32(a_r, br0,  accR0);   // + Ar*Br
    accR0 = wmma4_f32(a_i, nbi0, accR0);   // - Ai*Bi
    accI0 = wmma4_f32(a_r, bi0,  accI0);   // + Ar*Bi
    accI0 = wmma4_f32(a_i, br0,  accI0);   // + Ai*Br
    accR1 = wmma4_f32(a_r, br1,  accR1);
    accR1 = wmma4_f32(a_i, nbi1, accR1);
    accI1 = wmma4_f32(a_r, bi1,  accI1);
    accI1 = wmma4_f32(a_i, br1,  accI1);
  }

#pragma unroll
  for (int r = 0; r < 8; ++r) {
    const int row = tm + r + 8 * half;
    if (row < M) {
      const size_t ro = (size_t)row * rowsz;
      const int c0 = tn + l16;
      const int c1 = tn + 16 + l16;
      if (c0 < N) {
        Cr[ro + c0] = alpha * accR0[r];
        Ci[ro + c0] = alpha * accI0[r];
      }
      if (c1 < N) {
        Cr[ro + c1] = alpha * accR1[r];
        Ci[ro + c1] = alpha * accI1[r];
      }
    }
  }
}

// ---------------------------------------------------------------------------
// Twiddle generator: re/im[r*cols+c] = exp(sign * 2pi*i * ((r+roff)*(c+coff) % modn)/modn)
// Exact integer argument reduction keeps __sincosf accurate.
// ---------------------------------------------------------------------------
__global__ void gen_dft(float* __restrict__ re, float* __restrict__ im,
                        int rows, int cols, int roff, int coff, int modn, float sign)
{
  const int i = blockIdx.x * blockDim.x + threadIdx.x;
  if (i >= rows * cols) return;
  const int r = i / cols, c = i % cols;
  const int prod = ((r + roff) * (c + coff)) % modn;
  const float ang = sign * TWO_PI_F * (float)prod / (float)modn;
  float s, cc;
  __sincosf(ang, &s, &cc);
  re[i] = cc;
  im[i] = s;
}

// field = B1 * mask_real * exp(i*mask_param)  (B1 interleaved complex64)
__global__ void gen_field(const float* __restrict__ mask_param,
                          const float* __restrict__ mask_real,
                          const float* __restrict__ B1,
                          float* __restrict__ fr, float* __restrict__ fi)
{
  const int i = blockIdx.x * blockDim.x + threadIdx.x;
  if (i >= 500 * 500) return;
  float s, c;
  __sincosf(mask_param[i], &s, &c);
  const float mr  = mask_real[i];
  const float b1r = B1[2 * i], b1i = B1[2 * i + 1];
  fr[i] = mr * (b1r * c - b1i * s);
  fi[i] = mr * (b1r * s + b1i * c);
}

__global__ void deinterleave(const float* __restrict__ src,
                             float* __restrict__ re, float* __restrict__ im, int n)
{
  const int i = blockIdx.x * blockDim.x + threadIdx.x;
  if (i >= n) return;
  re[i] = src[2 * i];
  im[i] = src[2 * i + 1];
}

// In-place complex Hadamard: (ar,ai) *= (br,bi)
__global__ void cmul_inplace(float* __restrict__ ar, float* __restrict__ ai,
                             const float* __restrict__ br, const float* __restrict__ bi,
                             int n)
{
  const int i = blockIdx.x * blockDim.x + threadIdx.x;
  if (i >= n) return;
  const float xr = ar[i], xi = ai[i], yr = br[i], yi = bi[i];
  ar[i] = xr * yr - xi * yi;
  ai[i] = xr * yi + xi * yr;
}

// gshift = ifftshift(gamma_cust) for N=500 (even: index -> (idx+250)%500)
__global__ void shift_gamma(const float* __restrict__ g, float* __restrict__ gs)
{
  const int i = blockIdx.x * blockDim.x + threadIdx.x;
  if (i >= 500 * 500) return;
  const int k = i / 500, l = i % 500;
  gs[i] = g[((k + 250) % 500) * 500 + ((l + 250) % 500)];
}

// Column twiddles per emitter: tw[e,l] = exp(+2pi*i * c_e * l / 500), c_e = 249 + z_e
__global__ void gen_coltw(const int* __restrict__ xyz,
                          float* __restrict__ twr, float* __restrict__ twi)
{
  const int i = blockIdx.x * blockDim.x + threadIdx.x;
  if (i >= 128 * 500) return;
  const int e = i / 500, l = i % 500;
  int z = xyz[e * 3 + 2];
  if (z < 0) z = -z;
  const int c = 249 + z;
  const int prod = (c * l) % 500;
  const float ang = TWO_PI_F * (float)prod * (1.0f / 500.0f);
  float s, cc;
  __sincosf(ang, &s, &cc);
  twr[i] = cc;
  twi[i] = s;
}

__global__ void zero_kernel(float* __restrict__ p, int n)
{
  const int i = blockIdx.x * blockDim.x + threadIdx.x;
  if (i < n) p[i] = 0.0f;
}

// ---------------------------------------------------------------------------
// Per-emitter intensity: one wave per (e, k) row.
//   W = sum_l G[k,l] * exp(i*s*gshift[k,l]*xf_e) * tw[e,l]
//   intens[e] += |W|^2 / 500^3        (Parseval-collapsed column IFFT)
// ---------------------------------------------------------------------------
__global__ __launch_bounds__(128)
void emitter_kernel(const float* __restrict__ Gr, const float* __restrict__ Gi,
                    const float* __restrict__ gsh,
                    const float* __restrict__ twr, const float* __restrict__ twi,
                    const int* __restrict__ xyz,
                    float* __restrict__ intens)
{
  const int lane = threadIdx.x & 31;
  const int wave = threadIdx.x >> 5;
  const int k = blockIdx.x * 4 + wave;
  const int e = blockIdx.y;
  if (k >= 500) return;

  const float xf = (float)(xyz[e * 3 + 0] - 100);
  const float sconst = TWO_PI_F * 1.33f / 0.561f;  // K * 1e-6

  const float* gr = Gr  + (size_t)k * 500;
  const float* gi = Gi  + (size_t)k * 500;
  const float* gs = gsh + (size_t)k * 500;
  const float* tr = twr + (size_t)e * 500;
  const float* ti = twi + (size_t)e * 500;

  float wr = 0.0f, wi = 0.0f;
  for (int l = lane; l < 500; l += 32) {
    float sp, cp;
    __sincosf(sconst * gs[l] * xf, &sp, &cp);
    const float t_r = gr[l] * cp - gi[l] * sp;
    const float t_i = gr[l] * sp + gi[l] * cp;
    wr += t_r * tr[l] - t_i * ti[l];
    wi += t_r * ti[l] + t_i * tr[l];
  }
  for (int off = 16; off > 0; off >>= 1) {
    wr += __shfl_xor(wr, off, 32);
    wi += __shfl_xor(wi, off, 32);
  }
  if (lane == 0) {
    atomicAdd(&intens[e], (wr * wr + wi * wi) * (1.0f / 1.25e8f)); // 1/500^3
  }
}

// Scatter 31x31 psf patch * intensity into canvas (one block per emitter)
__global__ void scatter_kernel(const float* __restrict__ psf,
                               const int* __restrict__ xyz,
                               const float* __restrict__ intens,
                               float* __restrict__ canvas)
{
  const int e = blockIdx.x;          // 0..127, b = e/64
  const int b = e >> 6;
  const int x = xyz[e * 3 + 0];
  const int y = xyz[e * 3 + 1];
  int z = xyz[e * 3 + 2];
  if (z < 0) z = -z;
  const float inten = intens[e];
  const float* p = psf + (size_t)z * 961;
  float* cb = canvas + (size_t)b * 40000;
  for (int t = threadIdx.x; t < 961; t += blockDim.x) {
    const int dr = t / 31, dc = t % 31;
    atomicAdd(&cb[(x - 15 + dr) * 200 + (y - 15 + dc)], p[t] * inten);
  }
}

// Noise pipeline + normalization
__global__ void noise_kernel(const float* __restrict__ canvas,
                             const float* __restrict__ ng,
                             const float* __restrict__ npn,
                             float* __restrict__ out, int n)
{
  const int i = blockIdx.x * blockDim.x + threadIdx.x;
  if (i >= n) return;
  float inp = canvas[i] * (1.0f / 50000.0f) + 100.0f + 100000.0f
            + (2.0e8f * ng[i] + 3.0e8f);
  inp = (inp <= 0.0f) ? 0.0f : inp;
  float noisy = inp + 100.0f * sqrtf(inp) * npn[i];
  noisy = (noisy <= 10.0f) ? 1.0f : noisy;
  noisy = fminf(noisy, 4.0e9f);
  out[i] = noisy * (1.0f / 4.0e9f);
}

// ---------------------------------------------------------------------------
extern "C" void kernel_launch(void* const* d_in, const int* in_sizes, int n_in,
                              void* d_out, int out_size, void* d_ws, size_t ws_size,
                              hipStream_t stream)
{
  (void)in_sizes; (void)n_in; (void)out_size; (void)ws_size;

  const float* mask_param = (const float*)d_in[0];
  const int*   xyz        = (const int*)d_in[1];
  // d_in[2] = nphotons (unused by the reference forward pass)
  const float* B1         = (const float*)d_in[3];  // complex64 interleaved
  const float* Q1         = (const float*)d_in[4];  // complex64 interleaved
  const float* mask_real  = (const float*)d_in[5];
  const float* gamma_c    = (const float*)d_in[6];
  const float* psf        = (const float*)d_in[7];
  const float* ng         = (const float*)d_in[8];
  const float* npn        = (const float*)d_in[9];
  float* out = (float*)d_out;

  // ----- workspace layout (floats) -----
  float* w = (float*)d_ws;
  size_t off = 0;
  auto A = [&](size_t n) { float* p = w + off; off += n; return p; };
  float *fieldR = A(250000), *fieldI = A(250000);
  float *DpadR  = A(500000), *DpadI  = A(500000);   // 500x1000
  float *Rr     = A(500000), *Ri     = A(500000);   // 500x1000
  float *DpTr   = A(500000), *DpTi   = A(500000);   // 1000x500
  float *FEr    = A(1000000), *FEi   = A(1000000);  // 1000x1000 (becomes P)
  float *D1kr   = A(1000000), *D1ki  = A(1000000);  // 1000x1000
  float *Q1r    = A(1000000), *Q1i   = A(1000000);
  float *QRr    = A(1000000), *QRi   = A(1000000);
  float *FQr    = A(1000000), *FQi   = A(1000000);
  float *EinvR  = A(500000), *EinvI  = A(500000);   // 1000x500
  float *Sr     = A(500000), *Si     = A(500000);   // 1000x500
  float *EiTr   = A(500000), *EiTi   = A(500000);   // 500x1000
  float *outR   = A(250000), *outI   = A(250000);
  float *D5r    = A(250000), *D5i    = A(250000);
  float *G1r    = A(250000), *G1i    = A(250000);
  float *Gr     = A(250000), *Gi     = A(250000);
  float *gsh    = A(250000);
  float *twr    = A(64000),  *twi    = A(64000);
  float *intens = A(128);
  float *canvas = A(80000);

  const int T = 256;
  auto nb = [](int n, int t) { return (n + t - 1) / t; };
  // 32x64 macro tile per block
  auto gg = [](int M, int N) { return dim3((unsigned)((N + 63) / 64),
                                           (unsigned)((M + 31) / 32)); };

  // ----- build operands -----
  gen_field<<<nb(250000, T), T, 0, stream>>>(mask_param, mask_real, B1, fieldR, fieldI);
  deinterleave<<<nb(1000000, T), T, 0, stream>>>(Q1, Q1r, Q1i, 1000000);

  gen_dft<<<nb(500000, T), T, 0, stream>>>(DpadR, DpadI, 500, 1000, 250, 0, 1000, -1.0f);
  gen_dft<<<nb(500000, T), T, 0, stream>>>(DpTr,  DpTi,  1000, 500, 0, 250, 1000, -1.0f);
  gen_dft<<<nb(1000000, T), T, 0, stream>>>(D1kr, D1ki, 1000, 1000, 0, 0, 1000, -1.0f);
  gen_dft<<<nb(500000, T), T, 0, stream>>>(EinvR, EinvI, 1000, 500, 0, 750, 1000, 1.0f);
  gen_dft<<<nb(500000, T), T, 0, stream>>>(EiTr,  EiTi,  500, 1000, 750, 0, 1000, 1.0f);
  gen_dft<<<nb(250000, T), T, 0, stream>>>(D5r, D5i, 500, 500, 0, 0, 500, -1.0f);

  shift_gamma<<<nb(250000, T), T, 0, stream>>>(gamma_c, gsh);
  gen_coltw<<<nb(64000, T), T, 0, stream>>>(xyz, twr, twi);
  zero_kernel<<<1, 128, 0, stream>>>(intens, 128);
  zero_kernel<<<nb(80000, T), T, 0, stream>>>(canvas, 80000);

  // ----- FFTs as complex WMMA GEMMs -----
  // R = field @ Dpad                                  (500x1000, K=500)
  cgemm_wmma_f32<<<gg(500, 1000), 128, 0, stream>>>(fieldR, fieldI, DpadR, DpadI,
                                                    Rr, Ri, 500, 1000, 500, 1.0f);
  // FE = Dpad^T @ R = fft2(padded field)              (1000x1000, K=500)
  cgemm_wmma_f32<<<gg(1000, 1000), 128, 0, stream>>>(DpTr, DpTi, Rr, Ri,
                                                     FEr, FEi, 1000, 1000, 500, 1.0f);
  // QR = Q1 @ D1000 ; FQ = D1000 @ QR = fft2(Q1)      (1000x1000, K=1000)
  cgemm_wmma_f32<<<gg(1000, 1000), 128, 0, stream>>>(Q1r, Q1i, D1kr, D1ki,
                                                     QRr, QRi, 1000, 1000, 1000, 1.0f);
  cgemm_wmma_f32<<<gg(1000, 1000), 128, 0, stream>>>(D1kr, D1ki, QRr, QRi,
                                                     FQr, FQi, 1000, 1000, 1000, 1.0f);
  // P = FE .* FQ (in place into FE)
  cmul_inplace<<<nb(1000000, T), T, 0, stream>>>(FEr, FEi, FQr, FQi, 1000000);
  // S = P @ Einv                                      (1000x500, K=1000)
  cgemm_wmma_f32<<<gg(1000, 500), 128, 0, stream>>>(FEr, FEi, EinvR, EinvI,
                                                    Sr, Si, 1000, 500, 1000, 1.0f);
  // out = 1e-6 * Einv^T @ S  (cropped ifftshift(ifft2))  (500x500, K=1000)
  cgemm_wmma_f32<<<gg(500, 500), 128, 0, stream>>>(EiTr, EiTi, Sr, Si,
                                                   outR, outI, 500, 500, 1000, 1.0e-6f);
  // G = D500 @ out @ D500 = fft2(out)                 (500x500, K=500 twice)
  cgemm_wmma_f32<<<gg(500, 500), 128, 0, stream>>>(D5r, D5i, outR, outI,
                                                   G1r, G1i, 500, 500, 500, 1.0f);
  cgemm_wmma_f32<<<gg(500, 500), 128, 0, stream>>>(G1r, G1i, D5r, D5i,
                                                   Gr, Gi, 500, 500, 500, 1.0f);

  // ----- emitters, scatter, noise -----
  emitter_kernel<<<dim3(125, 128), 128, 0, stream>>>(Gr, Gi, gsh, twr, twi, xyz, intens);
  scatter_kernel<<<128, 256, 0, stream>>>(psf, xyz, intens, canvas);
  noise_kernel<<<nb(80000, T), T, 0, stream>>>(canvas, ng, npn, out, 80000);
}